// DualEncoderModel_81956565943065
// MI455X (gfx1250) — compile-verified
//
#include <hip/hip_runtime.h>
#include <hip/hip_bf16.h>

// Problem constants (from the reference)
#define B_   64
#define L_   50
#define A_   64
#define F_   4
#define E_   256
#define P_   1024
#define NF_  32          // A_/2
#define KF_  (L_ * F_)   // 200 flattened trajectory features per agent

typedef float v2f __attribute__((ext_vector_type(2)));
typedef float v8f __attribute__((ext_vector_type(8)));

// ---------------------------------------------------------------------------
// Kernel 1: per-agent encoder.  One block = one (batch, side, agent), thread = e.
// x[200] staged in LDS (broadcast), W reads coalesced across e.
// emb layout in workspace: [side][b][agent][e]  (side 0 = friendly, 1 = unauth)
// ---------------------------------------------------------------------------
__global__ __launch_bounds__(256) void encode_kernel(
    const float* __restrict__ traj,
    const float* __restrict__ Wf, const float* __restrict__ bf,
    const float* __restrict__ Wu, const float* __restrict__ bu,
    float* __restrict__ emb)
{
    __shared__ float x[KF_];
    const int blk  = blockIdx.x;        // [0, B*2*NF)
    const int b    = blk >> 6;
    const int sa   = blk & 63;          // side*NF + agent
    const int side = sa >> 5;
    const int ag   = sa & 31;
    const int agg  = side * NF_ + ag;   // global agent index
    const int e    = threadIdx.x;

    if (threadIdx.x < KF_) {
        const int l = threadIdx.x >> 2;       // k = l*F + f, F = 4
        const int f = threadIdx.x & 3;
        x[threadIdx.x] = traj[((b * L_ + l) * A_ + agg) * F_ + f];
    }
    __syncthreads();

    const float* W    = side ? Wu : Wf;
    const float* bias = side ? bu : bf;
    float acc = bias[e];
#pragma unroll 8
    for (int k = 0; k < KF_; ++k)
        acc = fmaf(x[k], W[k * E_ + e], acc);

    emb[((side * B_ + b) * NF_ + ag) * E_ + e] = acc;
}

// ---------------------------------------------------------------------------
// Kernel 2: pair MLP via fp32 WMMA.
// Block = (batch, 64-pair tile), 8 wave32.  Wave w owns ALL 64 rows and a
// 32-column slab (cols [w*32, w*32+32)) of h = relu(rel @ W1 + b1):
// 4 row-tiles x 2 col-tiles = 8 WMMA accumulator tiles (64 VGPRs).
// Each B fragment (2 global loads) feeds 4 WMMAs -> 0.5 loads/WMMA.
// rel is generated on the fly from ef/eu LDS chunks (4 segments: ef, eu,
// |ef-eu|, ef*eu).  Epilogue fuses bias + ReLU + W2 dot + mask.
// ---------------------------------------------------------------------------
__global__ __launch_bounds__(256) void pair_mlp_kernel(
    const float* __restrict__ emb,
    const int*   __restrict__ pairs,
    const unsigned char* __restrict__ mask,
    const float* __restrict__ W1, const float* __restrict__ b1,
    const float* __restrict__ W2, const float* __restrict__ b2p,
    float* __restrict__ out)
{
    constexpr int ROWS = 64;        // pairs per block
    constexpr int JC   = 64;        // j-chunk width (of E=256)
    constexpr int LDW  = JC + 4;    // padded LDS row stride (68 dwords)

    __shared__ float ef_t[ROWS * LDW];
    __shared__ float eu_t[ROWS * LDW];
    __shared__ int   f_off[ROWS];
    __shared__ int   u_off[ROWS];
    __shared__ float lds_logit[ROWS];

    const int b   = blockIdx.x >> 4;
    const int p0  = (blockIdx.x & 15) * ROWS;
    const int tid = threadIdx.x;

    const float* emb_f = emb;                      // [B][NF][E]
    const float* emb_u = emb + (size_t)B_ * NF_ * E_;

    if (tid < ROWS) {
        const int fi = pairs[(b * P_ + p0 + tid) * 2 + 0];        // [0, NF)
        const int ui = pairs[(b * P_ + p0 + tid) * 2 + 1] - NF_;  // [0, NF)
        f_off[tid] = (b * NF_ + fi) * E_;
        u_off[tid] = (b * NF_ + ui) * E_;
        lds_logit[tid] = 0.0f;
    }
    __syncthreads();

    const int lane    = tid & 31;
    const int w       = tid >> 5;         // wave id = column slab 0..7
    const int colbase = w * 32;
    const int n       = lane & 15;        // N column within 16-wide tile
    const int half    = lane >> 4;        // 0: lanes 0-15, 1: lanes 16-31
    const int kh      = half << 1;        // K half: 0 -> K0/K1, 1 -> K2/K3

    v8f acc[4][2];                        // [row-tile][col-tile]
#pragma unroll
    for (int rt = 0; rt < 4; ++rt)
#pragma unroll
        for (int ct = 0; ct < 2; ++ct) { v8f z = {}; acc[rt][ct] = z; }

    for (int jc = 0; jc < 4; ++jc) {
        __syncthreads();
        // Gather ef/eu chunk [64 rows x 64 j] into LDS with float4 loads.
        for (int idx = tid; idx < ROWS * 16; idx += 256) {
            const int r  = idx >> 4;
            const int jv = (idx & 15) * 4;
            const float4 vf = *(const float4*)(emb_f + f_off[r] + jc * JC + jv);
            const float4 vu = *(const float4*)(emb_u + u_off[r] + jc * JC + jv);
            float* df = &ef_t[r * LDW + jv];
            float* du = &eu_t[r * LDW + jv];
            df[0] = vf.x; df[1] = vf.y; df[2] = vf.z; df[3] = vf.w;
            du[0] = vu.x; du[1] = vu.y; du[2] = vu.z; du[3] = vu.w;
        }
        __syncthreads();

        for (int kk = 0; kk < JC; kk += 4) {
            // A fragments for all 4 row tiles, 4 rel segments each.
            v2f afr[4][4];                // [row-tile][segment]
#pragma unroll
            for (int rt = 0; rt < 4; ++rt) {
                const int arow = rt * 16 + n;     // lanes 0-15 and 16-31 both hold M=0..15
                const float ef0 = ef_t[arow * LDW + kk + kh];
                const float ef1 = ef_t[arow * LDW + kk + kh + 1];
                const float eu0 = eu_t[arow * LDW + kk + kh];
                const float eu1 = eu_t[arow * LDW + kk + kh + 1];
                v2f a0 = {ef0, ef1};                                  afr[rt][0] = a0;
                v2f a1 = {eu0, eu1};                                  afr[rt][1] = a1;
                v2f a2 = {fabsf(ef0 - eu0), fabsf(ef1 - eu1)};        afr[rt][2] = a2;
                v2f a3 = {ef0 * eu0, ef1 * eu1};                      afr[rt][3] = a3;
            }

#pragma unroll
            for (int s = 0; s < 4; ++s) {
                const int krow = s * E_ + jc * JC + kk + kh;  // W1 row for this lane's K half
                const float* bp = W1 + (size_t)krow * E_ + colbase + n;
                v2f bfr0 = {bp[0],  bp[E_]};        // col-tile 0: K rows krow, krow+1
                v2f bfr1 = {bp[16], bp[E_ + 16]};   // col-tile 1
#pragma unroll
                for (int rt = 0; rt < 4; ++rt) {
                    acc[rt][0] = __builtin_amdgcn_wmma_f32_16x16x4_f32(
                        false, afr[rt][s], false, bfr0, (short)0, acc[rt][0], false, false);
                    acc[rt][1] = __builtin_amdgcn_wmma_f32_16x16x4_f32(
                        false, afr[rt][s], false, bfr1, (short)0, acc[rt][1], false, false);
                }
            }
        }
    }

    // Epilogue: h = relu(acc + b1); partial logits = h * W2, reduced per row.
    float part[4][8];
#pragma unroll
    for (int rt = 0; rt < 4; ++rt)
#pragma unroll
        for (int i = 0; i < 8; ++i) part[rt][i] = 0.0f;

#pragma unroll
    for (int ct = 0; ct < 2; ++ct) {
        const int col  = colbase + ct * 16 + n;
        const float bb = b1[col];
        const float w2 = W2[col];
#pragma unroll
        for (int rt = 0; rt < 4; ++rt) {
#pragma unroll
            for (int i = 0; i < 8; ++i) {
                float h = acc[rt][ct][i] + bb;
                h = fmaxf(h, 0.0f);
                part[rt][i] = fmaf(h, w2, part[rt][i]);
            }
        }
    }
    // Reduce across the 16 N-lanes inside each lane-half (rows differ by half).
#pragma unroll
    for (int off = 1; off < 16; off <<= 1) {
#pragma unroll
        for (int rt = 0; rt < 4; ++rt)
#pragma unroll
            for (int i = 0; i < 8; ++i)
                part[rt][i] += __shfl_xor(part[rt][i], off, 32);
    }
    if (n == 0) {
        // C/D layout: VGPR i holds M=i (lanes 0-15) or M=i+8 (lanes 16-31).
#pragma unroll
        for (int rt = 0; rt < 4; ++rt)
#pragma unroll
            for (int i = 0; i < 8; ++i)
                atomicAdd(&lds_logit[rt * 16 + half * 8 + i], part[rt][i]);
    }
    __syncthreads();

    if (tid < ROWS) {
        const int p  = p0 + tid;
        const int i0 = pairs[(b * P_ + p) * 2 + 0];
        const int i1 = pairs[(b * P_ + p) * 2 + 1];
        const bool valid = (mask[b * A_ + i0] != 0) && (mask[b * A_ + i1] != 0);
        out[b * P_ + p] = valid ? (lds_logit[tid] + b2p[0]) : 0.0f;
    }
}

// ---------------------------------------------------------------------------
extern "C" void kernel_launch(void* const* d_in, const int* in_sizes, int n_in,
                              void* d_out, int out_size, void* d_ws, size_t ws_size,
                              hipStream_t stream)
{
    (void)in_sizes; (void)n_in; (void)out_size; (void)ws_size;

    const float*         traj  = (const float*)d_in[0];
    // d_in[1] = batch_roles (unused by the math)
    const int*           pairs = (const int*)d_in[2];
    const unsigned char* mask  = (const unsigned char*)d_in[3];
    const float*         Wf    = (const float*)d_in[4];
    const float*         bf    = (const float*)d_in[5];
    const float*         Wu    = (const float*)d_in[6];
    const float*         bu    = (const float*)d_in[7];
    const float*         W1    = (const float*)d_in[8];
    const float*         b1    = (const float*)d_in[9];
    const float*         W2    = (const float*)d_in[10];
    const float*         b2    = (const float*)d_in[11];

    float* emb = (float*)d_ws;           // 2 * B * NF * E floats = 4 MB
    float* out = (float*)d_out;          // B * P floats

    encode_kernel<<<B_ * 2 * NF_, 256, 0, stream>>>(traj, Wf, bf, Wu, bu, emb);
    pair_mlp_kernel<<<B_ * (P_ / 64), 256, 0, stream>>>(emb, pairs, mask,
                                                        W1, b1, W2, b2, out);
}